// SAFilterTorch_30683246362698
// MI455X (gfx1250) — compile-verified
//
#include <hip/hip_runtime.h>
#include <cmath>

// ---------------------------------------------------------------------------
// SA filter (double-exponential synapse), B=8, T=4096, N=2048, f32.
// Memory-bound (512MB min traffic @ 23.3 TB/s ~ 22us). Implemented as a
// chunk-parallel linear scan:
//   phase1: per-chunk forced end states (zero IC)          [read 256MB]
//   phase2: 2x2 state-matrix prefix combine across chunks  [8MB]
//   phase3: intra-chunk outputs via WMMA f32 16x16x4 tiles [read 256MB, write 256MB]
// Phase 3 runs in reverse stream order with NT stores to harvest L2 (192MB)
// reuse of the phase-1 input stream.
// ---------------------------------------------------------------------------

typedef float v2f __attribute__((ext_vector_type(2)));
typedef float v8f __attribute__((ext_vector_type(8)));

namespace {
constexpr int B = 8, T = 4096, N = 2048;
constexpr int CH = 64;   // number of time chunks
constexpr int CL = 64;   // chunk length (T / CH)
constexpr int TILE = 16; // WMMA tile height (time steps)

constexpr float DT = 0.1f;
constexpr float AR = 0.1f / 5.0f;   // dt/tau_r
constexpr float AD = 0.1f / 30.0f;  // dt/tau_d
constexpr float RD = 1.0f - AR;     // x decay per step
constexpr float DD = 1.0f - AD;     // i decay per step
constexpr float C1 = AR * (3.0f / DT + 0.05f); // u_t =  C1*I_t - C2*I_{t-1}
constexpr float C2 = AR * (3.0f / DT);
} // namespace

// Host-precomputed constant tables (double precision on CPU, passed by value).
struct Tab {
  float V[16];   // impulse-response matrix entries, index s = m-k
  float XC[16];  // x_end dot coefficients, index k
  float TM[16];  // T(m) = a_d * (d^{m+1}-r^{m+1})/(d-r)
  float DM[16];  // d^{m+1}
  float R16, R15C2; // tile-level state propagation constants
};

// ---------------------------------------------------------------------------
// Phase 1: per-chunk forced response end state (x_e, i_e), zero initial state.
// Thread handles 4 consecutive channels (float4), one chunk of 64 steps.
// ---------------------------------------------------------------------------
__global__ __launch_bounds__(256) void sa_phase1_states(
    const float* __restrict__ I, float* __restrict__ xs, float* __restrict__ is_)
{
  int tid = blockIdx.x * blockDim.x + threadIdx.x;
  int g4 = tid & (N / 4 - 1);
  int t2 = tid / (N / 4);
  int c  = t2 & (CH - 1);
  int b  = t2 >> 6;            // log2(CH) = 6
  int n  = g4 * 4;
  int t0 = c * CL;

  const float* p = I + ((size_t)b * T + t0) * N + n;
  float4 prev = (t0 == 0) ? *(const float4*)p : *(const float4*)(p - N);
  float4 x = {0.f, 0.f, 0.f, 0.f};
  float4 a = {0.f, 0.f, 0.f, 0.f};
#pragma unroll 4
  for (int t = 0; t < CL; ++t) {
    float4 cur = *(const float4*)p;
    __builtin_prefetch(p + 8 * N, 0, 1);  // global_prefetch_b8, stride = 1 row
    p += N;
    x.x = RD * x.x + C1 * cur.x - C2 * prev.x;
    x.y = RD * x.y + C1 * cur.y - C2 * prev.y;
    x.z = RD * x.z + C1 * cur.z - C2 * prev.z;
    x.w = RD * x.w + C1 * cur.w - C2 * prev.w;
    a.x = DD * a.x + AD * x.x;
    a.y = DD * a.y + AD * x.y;
    a.z = DD * a.z + AD * x.z;
    a.w = DD * a.w + AD * x.w;
    prev = cur;
  }
  size_t o = ((size_t)b * CH + c) * N + n;
  *(float4*)(xs + o)  = x;
  *(float4*)(is_ + o) = a;
}

// ---------------------------------------------------------------------------
// Phase 2: prefix-combine chunk states.  s_init[c] = M^64 s_init[c-1] + v[c-1]
// with M^64 = [[rL, 0], [m21, dL]] (precomputed on host). In-place rewrite of
// (xs, is) from forced-end-state to chunk-initial-state.
// ---------------------------------------------------------------------------
__global__ __launch_bounds__(256) void sa_phase2_combine(
    float* __restrict__ xs, float* __restrict__ is_, float rL, float dL, float m21)
{
  int tid = blockIdx.x * blockDim.x + threadIdx.x;
  int n = tid & (N - 1);
  int b = tid / N;
  float xr = 0.f, ir = 0.f;
  for (int c = 0; c < CH; ++c) {
    size_t o = ((size_t)b * CH + c) * N + n;
    float vx = xs[o], vi = is_[o];
    xs[o]  = xr;
    is_[o] = ir;
    float xn = rL * xr + vx;
    float in = dL * ir + m21 * xr + vi;
    xr = xn; ir = in;
  }
}

// ---------------------------------------------------------------------------
// Phase 3: outputs.  One wave handles (b, chunk, 16-channel group); each 16x16
// output tile (16 timesteps x 16 channels) computed as D = V * Itile + C_pre
// via 4 chained V_WMMA_F32_16X16X4_F32 (K = 16).
//   A (16x4 f32) layout: lane L holds row m = L&15, K = g + 2*(L>>4) (+4q).
//   B (4x16 f32) layout: lane L holds col n = L&15, K = g + 2*(L>>4) (+4q).
//   C/D layout: VGPR j -> row m = j + 8*(L>>4), col n = L&15.
// Cross-tile state via a length-16 dot product + cross-lane shuffles.
// ---------------------------------------------------------------------------
__global__ __launch_bounds__(128) void sa_phase3_wmma(
    const float* __restrict__ I, const float* __restrict__ xs,
    const float* __restrict__ is_, float* __restrict__ out, Tab tab)
{
  int lane = threadIdx.x & 31;
  int w = blockIdx.x * 4 + (threadIdx.x >> 5);
  w = (B * CH * (N / 16) - 1) - w;   // reverse order: re-read phase-1 tail from L2
  int g16 = w & (N / 16 - 1);
  int t2  = w / (N / 16);
  int c   = t2 & (CH - 1);
  int b   = t2 >> 6;

  int hb = lane >> 4;       // half-wave: K offset +2, C/D row offset +8
  int ln = lane & 15;
  int n  = g16 * 16 + ln;
  int t0 = c * CL;

  // Per-lane constants from host tables.
  v2f A[4], XC[4];
#pragma unroll
  for (int q = 0; q < 4; ++q) {
    int k0 = 4 * q + 2 * hb;
    int s0 = ln - k0, s1 = ln - (k0 + 1);
    A[q].x = (s0 >= 0) ? tab.V[s0] : 0.f;
    A[q].y = (s1 >= 0) ? tab.V[s1] : 0.f;
    XC[q].x = tab.XC[k0];
    XC[q].y = tab.XC[k0 + 1];
  }
  float TMr[8], DMr[8];
#pragma unroll
  for (int j = 0; j < 8; ++j) {
    int m = j + 8 * hb;
    TMr[j] = tab.TM[m];
    DMr[j] = tab.DM[m];
  }

  size_t so = ((size_t)b * CH + c) * N + n;
  float x0 = xs[so];
  float i0 = is_[so];
  const float* ibase = I + ((size_t)b * T + t0) * N + n;
  float iprev = (t0 == 0) ? ibase[0] : *(ibase - N);
  float* obase = out + ((size_t)b * T + t0) * N + n;

#pragma unroll
  for (int tile = 0; tile < CL / TILE; ++tile) {
    const float* tb = ibase + (size_t)(tile * TILE) * N;
    v2f bm[4];
#pragma unroll
    for (int q = 0; q < 4; ++q) {
      int k0 = 4 * q + 2 * hb;
      bm[q].x = tb[(size_t)k0 * N];
      bm[q].y = tb[(size_t)(k0 + 1) * N];
      __builtin_prefetch(tb + (size_t)(k0 + TILE) * N, 0, 1);
    }
    // C_pre: homogeneous response of (x0, i0) + boundary term from I_{-1}.
    float hx = RD * x0 - C2 * iprev;
    v8f acc;
#pragma unroll
    for (int j = 0; j < 8; ++j) acc[j] = TMr[j] * hx + DMr[j] * i0;
    // D = V * Itile + C_pre : chain 4 x wmma f32 16x16x4 over K.
#pragma unroll
    for (int q = 0; q < 4; ++q)
      acc = __builtin_amdgcn_wmma_f32_16x16x4_f32(
          false, A[q], false, bm[q], (short)0, acc, false, false);
    // x_end = r^16 x0 - C2 r^15 I_{-1} + sum_k XC[k] * I_k  (split across halves)
    float px = 0.f;
#pragma unroll
    for (int q = 0; q < 4; ++q)
      px = fmaf(XC[q].x, bm[q].x, fmaf(XC[q].y, bm[q].y, px));
    px += __shfl_xor(px, 16, 32);
    float xend = tab.R16 * x0 - tab.R15C2 * iprev + px;
    float iend = __shfl(acc[7], 16 + ln, 32);   // row m=15 lives in lanes 16..31, VGPR 7
    float inew = __shfl(bm[3].y, 16 + ln, 32);  // I row 15 lives in lanes 16..31, q=3,g=1
    // Store 16x16 output tile (row m = j + 8*hb), non-temporal.
    float* op = obase + (size_t)(tile * TILE + 8 * hb) * N;
#pragma unroll
    for (int j = 0; j < 8; ++j)
      __builtin_nontemporal_store(acc[j], op + (size_t)j * N);
    x0 = xend; i0 = iend; iprev = inew;
  }
}

// ---------------------------------------------------------------------------
// Fallback (ws too small): plain sequential scan, thread = 4 channels.
// ---------------------------------------------------------------------------
__global__ __launch_bounds__(256) void sa_fallback(
    const float* __restrict__ I, float* __restrict__ out)
{
  int tid = blockIdx.x * blockDim.x + threadIdx.x;
  if (tid >= B * (N / 4)) return;
  int g4 = tid & (N / 4 - 1);
  int b  = tid / (N / 4);
  int n  = g4 * 4;
  const float* p = I + (size_t)b * T * N + n;
  float* o = out + (size_t)b * T * N + n;
  float4 prev = *(const float4*)p;
  float4 x = {0.f, 0.f, 0.f, 0.f};
  float4 a = {0.f, 0.f, 0.f, 0.f};
  for (int t = 0; t < T; ++t) {
    float4 cur = *(const float4*)p;
    p += N;
    x.x = RD * x.x + C1 * cur.x - C2 * prev.x;
    x.y = RD * x.y + C1 * cur.y - C2 * prev.y;
    x.z = RD * x.z + C1 * cur.z - C2 * prev.z;
    x.w = RD * x.w + C1 * cur.w - C2 * prev.w;
    a.x = DD * a.x + AD * x.x;
    a.y = DD * a.y + AD * x.y;
    a.z = DD * a.z + AD * x.z;
    a.w = DD * a.w + AD * x.w;
    prev = cur;
    *(float4*)o = a;
    o += N;
  }
}

// ---------------------------------------------------------------------------
extern "C" void kernel_launch(void* const* d_in, const int* in_sizes, int n_in,
                              void* d_out, int out_size, void* d_ws, size_t ws_size,
                              hipStream_t stream) {
  (void)in_sizes; (void)n_in; (void)out_size;
  const float* I = (const float*)d_in[0];
  float* out = (float*)d_out;

  const size_t nstate = (size_t)B * CH * N;
  const size_t need = 2 * nstate * sizeof(float); // 8 MB
  if (ws_size >= need) {
    float* xs  = (float*)d_ws;
    float* is_ = xs + nstate;

    // Host-side double-precision constant tables (deterministic).
    const double r  = 1.0 - 0.1 / 5.0;
    const double d  = 1.0 - 0.1 / 30.0;
    const double ad = 0.1 / 30.0;
    const double c1 = (0.1 / 5.0) * (3.0 / 0.1 + 0.05);
    const double c2 = (0.1 / 5.0) * (3.0 / 0.1);
    auto Tf = [&](int s) -> double {
      if (s < 0) return 0.0;
      return ad * (std::pow(d, s + 1) - std::pow(r, s + 1)) / (d - r);
    };
    Tab tab;
    for (int s = 0; s < 16; ++s)
      tab.V[s] = (float)(c1 * Tf(s) - c2 * Tf(s - 1));
    for (int k = 0; k < 16; ++k)
      tab.XC[k] = (float)(c1 * std::pow(r, 15 - k) -
                          (k < 15 ? c2 * std::pow(r, 14 - k) : 0.0));
    for (int m = 0; m < 16; ++m) {
      tab.TM[m] = (float)Tf(m);
      tab.DM[m] = (float)std::pow(d, m + 1);
    }
    tab.R16   = (float)std::pow(r, 16);
    tab.R15C2 = (float)(c2 * std::pow(r, 15));
    const float rL  = (float)std::pow(r, CL);
    const float dL  = (float)std::pow(d, CL);
    const float m21 = (float)(r * Tf(CL - 1));

    sa_phase1_states<<<B * CH * (N / 4) / 256, 256, 0, stream>>>(I, xs, is_);
    sa_phase2_combine<<<B * N / 256, 256, 0, stream>>>(xs, is_, rL, dL, m21);
    sa_phase3_wmma<<<B * CH * (N / 16) / 4, 128, 0, stream>>>(I, xs, is_, out, tab);
  } else {
    sa_fallback<<<(B * (N / 4) + 255) / 256, 256, 0, stream>>>(I, out);
  }
}